// FCOS_78168404787576
// MI455X (gfx1250) — compile-verified
//
#include <hip/hip_runtime.h>
#include <hip/hip_bf16.h>
#include <math.h>

#define INF_A   100000000.0f
#define ALPHA_P 0.25f
#define ALPHA_N 0.75f
#define RADIUS_ 1.5f
#define NTHREADS 256
#define NWAVES   (NTHREADS / 32)

typedef float v2f __attribute__((ext_vector_type(2)));
typedef float v8f __attribute__((ext_vector_type(8)));

// ---------------------------------------------------------------------------
// Wave-wide f32 sum using V_WMMA_F32_16X16X4_F32.
// A: lane's partial in its first K-slot, 0 in the second; B: all ones.
// D[m,n] = rowsum_m(A). Each lane's 8 D components are 8 distinct row-sums;
// lane n and lane n^16 together cover all 16 rows exactly once, so
// (sum of 8 components) + shfl_xor(16) == sum over all 32 lanes, exact f32.
// Must be called with full EXEC (all threads of the wave converged).
// ---------------------------------------------------------------------------
__device__ __forceinline__ float wave_sum_wmma(float v) {
    v2f a;  a[0] = v;    a[1] = 0.0f;
    v2f o;  o[0] = 1.0f; o[1] = 1.0f;
    v8f c = {0.0f, 0.0f, 0.0f, 0.0f, 0.0f, 0.0f, 0.0f, 0.0f};
    c = __builtin_amdgcn_wmma_f32_16x16x4_f32(false, a, false, o,
                                              (short)0, c, false, false);
    float s = c[0] + c[1] + c[2] + c[3] + c[4] + c[5] + c[6] + c[7];
    s += __shfl_xor(s, 16, 32);
    return s;
}

// stable focal-loss term for one logit
__device__ __forceinline__ float focal_term(float x, bool is_pos) {
    float e    = expf(-fabsf(x));
    float l1pe = log1pf(e);
    float lsx  = fminf(x, 0.0f) - l1pe;    // log sigmoid(x)
    float lsn  = fminf(-x, 0.0f) - l1pe;   // log sigmoid(-x)
    float t    = 1.0f / (1.0f + e);
    float prb  = (x >= 0.0f) ? t : (1.0f - t);  // sigmoid(x)
    float onem = 1.0f - prb;
    return is_pos ? (-ALPHA_P * onem * onem * lsx)
                  : (-ALPHA_N * prb * prb * lsn);
}

// ---------------------------------------------------------------------------
// Main fused kernel: assignment + focal + GIoU + centerness, one (b,p)/thread.
// Writes 5 per-block partial sums to ws[blockIdx*5 + {fl,npos,cent,boxnum,cen}]
// ---------------------------------------------------------------------------
__global__ __launch_bounds__(NTHREADS)
void fcos_main(const float* __restrict__ loc,  const float* __restrict__ cls,
               const float* __restrict__ boxp, const float* __restrict__ cenp,
               const float* __restrict__ gtb,  const int*   __restrict__ gtl,
               float* __restrict__ ws, int P, int C, int G, int pblk)
{
    __shared__ float sx1[128], sy1[128], sx2[128], sy2[128];
    __shared__ float scx[128], scy[128], sar[128];
    __shared__ int   slb[128];
    __shared__ float sred[NWAVES][5];

    const int tid = threadIdx.x;
    const int bb  = blockIdx.x / pblk;   // batch
    const int pb  = blockIdx.x % pblk;   // point-chunk
    const int p   = pb * NTHREADS + tid;

    if (tid < G) {
        const float4 bx = reinterpret_cast<const float4*>(gtb)[(size_t)bb * G + tid];
        sx1[tid] = bx.x; sy1[tid] = bx.y; sx2[tid] = bx.z; sy2[tid] = bx.w;
        scx[tid] = (bx.x + bx.z) * 0.5f;
        scy[tid] = (bx.y + bx.w) * 0.5f;
        sar[tid] = (bx.z - bx.x) * (bx.w - bx.y);
        slb[tid] = gtl[(size_t)bb * G + tid];
    }
    __syncthreads();

    float fl_s = 0.0f, np_s = 0.0f, ct_s = 0.0f, bn_s = 0.0f, cl_s = 0.0f;

    if (p < P) {
        const float xs = loc[2 * p], ys = loc[2 * p + 1];

        // per-level constants (STRIDES/FEAT/SIZES from reference)
        float sp, lo, hi;
        if      (p < 10000) { sp =   8.0f; lo =  -1.0f; hi =  64.0f; }
        else if (p < 12500) { sp =  16.0f; lo =  64.0f; hi = 128.0f; }
        else if (p < 13125) { sp =  32.0f; lo = 128.0f; hi = 256.0f; }
        else if (p < 13294) { sp =  64.0f; lo = 256.0f; hi = 512.0f; }
        else                { sp = 128.0f; lo = 512.0f; hi = INF_A;  }
        const float rad = sp * RADIUS_;

        // ---- assignment: first-occurrence argmin of masked area ----
        float best_a = INF_A; int best_g = 0;
        for (int g = 0; g < G; ++g) {
            float x1 = sx1[g], y1 = sy1[g], x2 = sx2[g], y2 = sy2[g];
            float l  = xs - x1, t = ys - y1, r = x2 - xs, b = y2 - ys;
            float cx = scx[g], cy = scy[g];
            float cgx1 = fmaxf(cx - rad, x1), cgy1 = fmaxf(cy - rad, y1);
            float cgx2 = fminf(cx + rad, x2), cgy2 = fminf(cy + rad, y2);
            float inb  = fminf(fminf(xs - cgx1, cgx2 - xs),
                               fminf(ys - cgy1, cgy2 - ys));
            float mx   = fmaxf(fmaxf(l, t), fmaxf(r, b));
            bool  ok   = (inb > 0.0f) && (mx >= lo) && (mx <= hi);
            float a    = ok ? sar[g] : INF_A;
            if (a < best_a) { best_a = a; best_g = g; }
        }
        const float tl = xs - sx1[best_g];
        const float tt = ys - sy1[best_g];
        const float tr = sx2[best_g] - xs;
        const float tb = sy2[best_g] - ys;
        const int   lab = (best_a >= INF_A) ? 0 : slb[best_g];
        const bool  pos = lab > 0;

        // ---- centerness target ----
        float lr_min = fminf(tl, tr), lr_max = fmaxf(tl, tr);
        float tb_min = fminf(tt, tb), tb_max = fmaxf(tt, tb);
        float cent = sqrtf(fmaxf((lr_min / fmaxf(lr_max, 1e-7f)) *
                                 (tb_min / fmaxf(tb_max, 1e-7f)), 0.0f));
        if (!pos) cent = 0.0f;

        // ---- GIoU loss ----
        const size_t row = (size_t)bb * P + p;
        const float4 bp  = reinterpret_cast<const float4*>(boxp)[row];
        const float pl = bp.x, pt = bp.y, pr = bp.z, pb_ = bp.w;
        float t_area = (tl + tr) * (tt + tb);
        float p_area = (pl + pr) * (pt + pb_);
        float ai = (fminf(pl, tl) + fminf(pr, tr)) * (fminf(pb_, tb) + fminf(pt, tt));
        float au = t_area + p_area - ai;
        float ious = (ai + 1.0f) / (au + 1.0f);
        float gi = (fmaxf(pl, tl) + fmaxf(pr, tr)) * (fmaxf(pb_, tb) + fmaxf(pt, tt)) + 1e-7f;
        float giou_loss = 1.0f - (ious - (gi - au) / gi);

        // ---- centerness BCE ----
        float xc  = cenp[row];
        float bce = fmaxf(xc, 0.0f) - xc * cent + log1pf(expf(-fabsf(xc)));

        // ---- focal loss over C classes (float4 loads) ----
        const float4* crow = reinterpret_cast<const float4*>(cls + row * (size_t)C);
        float fl = 0.0f;
        const int c4n = C >> 2;
        for (int i = 0; i < c4n; ++i) {
            float4 v = crow[i];
            fl += focal_term(v.x, lab == 4 * i + 1);
            fl += focal_term(v.y, lab == 4 * i + 2);
            fl += focal_term(v.z, lab == 4 * i + 3);
            fl += focal_term(v.w, lab == 4 * i + 4);
        }
        for (int cc = c4n * 4; cc < C; ++cc)
            fl += focal_term(cls[row * (size_t)C + cc], lab == cc + 1);

        fl_s = fl;
        np_s = pos ? 1.0f : 0.0f;
        ct_s = cent;
        bn_s = giou_loss * cent;      // cent==0 for negatives, as in reference
        cl_s = pos ? bce : 0.0f;
    }

    // ---- deterministic block reduction (WMMA per wave, serial across waves)
    float q[5] = {fl_s, np_s, ct_s, bn_s, cl_s};
    const int wid = tid >> 5, lane = tid & 31;
    #pragma unroll
    for (int i = 0; i < 5; ++i) {
        float w = wave_sum_wmma(q[i]);
        if (lane == 0) sred[wid][i] = w;
    }
    __syncthreads();
    if (tid == 0) {
        #pragma unroll
        for (int i = 0; i < 5; ++i) {
            float s = 0.0f;
            for (int w = 0; w < NWAVES; ++w) s += sred[w][i];
            ws[(size_t)blockIdx.x * 5 + i] = s;
        }
    }
}

// ---------------------------------------------------------------------------
// Finalize: one block reduces per-block partials (fixed order -> deterministic)
// and emits [cls_loss, box_loss, center_loss].
// ---------------------------------------------------------------------------
__global__ __launch_bounds__(NTHREADS)
void fcos_finalize(const float* __restrict__ ws, int nrows, float Bf,
                   float* __restrict__ out)
{
    __shared__ float sred[NWAVES][5];
    const int tid = threadIdx.x;
    float acc[5] = {0.0f, 0.0f, 0.0f, 0.0f, 0.0f};
    for (int r = tid; r < nrows; r += NTHREADS) {
        #pragma unroll
        for (int i = 0; i < 5; ++i) acc[i] += ws[(size_t)r * 5 + i];
    }
    const int wid = tid >> 5, lane = tid & 31;
    #pragma unroll
    for (int i = 0; i < 5; ++i) {
        float w = wave_sum_wmma(acc[i]);
        if (lane == 0) sred[wid][i] = w;
    }
    __syncthreads();
    if (tid == 0) {
        float tot[5];
        #pragma unroll
        for (int i = 0; i < 5; ++i) {
            float s = 0.0f;
            for (int w = 0; w < NWAVES; ++w) s += sred[w][i];
            tot[i] = s;
        }
        const float fl = tot[0], np = tot[1], ct = tot[2], bn = tot[3], cl = tot[4];
        out[0] = fl / (np + Bf);               // cls_loss
        out[1] = bn / fmaxf(ct, 1e-6f);        // box_loss
        out[2] = cl / fmaxf(np, 1.0f);         // center_loss
    }
}

extern "C" void kernel_launch(void* const* d_in, const int* in_sizes, int n_in,
                              void* d_out, int out_size, void* d_ws, size_t ws_size,
                              hipStream_t stream)
{
    (void)n_in; (void)out_size; (void)ws_size;
    const float* loc  = (const float*)d_in[0];   // (P,2)
    const float* cls  = (const float*)d_in[1];   // (B,P,C)
    const float* boxp = (const float*)d_in[2];   // (B,P,4)
    const float* cenp = (const float*)d_in[3];   // (B,P)
    const float* gtb  = (const float*)d_in[4];   // (B,G,4)
    const int*   gtl  = (const int*)d_in[5];     // (B,G)

    const int P  = in_sizes[0] / 2;
    const int Bn = in_sizes[3] / P;
    const int C  = in_sizes[1] / (Bn * P);
    const int G  = in_sizes[5] / Bn;

    const int pblk    = (P + NTHREADS - 1) / NTHREADS;
    const int nblocks = Bn * pblk;
    float* ws = (float*)d_ws;

    fcos_main<<<nblocks, NTHREADS, 0, stream>>>(loc, cls, boxp, cenp, gtb, gtl,
                                                ws, P, C, G, pblk);
    fcos_finalize<<<1, NTHREADS, 0, stream>>>(ws, nblocks, (float)Bn,
                                              (float*)d_out);
}